// Droppart_88347477279038
// MI455X (gfx1250) — compile-verified
//
#include <hip/hip_runtime.h>

// DropPart for x[64,256,96,32] fp32, GROUPS=8, P=0.5, NUM_KPTS=17.
// HBM-bound streaming select (~320 MB effective traffic -> ~14 us floor at
// 23.3 TB/s). Per-(n,g) uniform box params load via SMEM; rows provably
// all-zero under the gate skip their HBM reads via EXEC-predicated loads.
// GRIDX_=6 amortizes the ~150-instr block setup over 16 memory iterations so
// instruction issue (~6 us) stays well under the memory floor.
// No WMMA: the op has zero matmul FLOPs.

#define N_    64
#define C_    256
#define H_    96
#define W_    32
#define GROUPS_ 8
#define CHS_  (C_ / GROUPS_)          // 32 channels per group
#define NKPT_ 17
#define P_    0.5f
#define SHALF_ 8.0f                   // int(0.25 * W) = 8

#define GRIDX_ 6
#define F4_PER_ROW_ (W_ / 4)          // 8
#define ROWS_PER_TILE_ 32             // 256 threads / 8 float4-per-row
#define ROWS_PER_SLAB_ (CHS_ * H_)    // 3072
#define ROW_STRIDE_ (GRIDX_ * ROWS_PER_TILE_)               // 192 (multiple of H_!)
#define ITERS_ (ROWS_PER_SLAB_ / ROW_STRIDE_)               // 16
#define F4_STRIDE_ (ROW_STRIDE_ * F4_PER_ROW_)              // 1536

// Native clang vector: accepted by __builtin_nontemporal_*, lowers to
// global_load_b128 / global_store_b128 with TH_NT.
typedef float vf4 __attribute__((ext_vector_type(4)));

__global__ __launch_bounds__(256) void droppart_kernel(
    const float* __restrict__ x,
    const float* __restrict__ kpts,
    const float* __restrict__ roll,
    float* __restrict__ out)
{
    const int n = blockIdx.z;
    const int g = blockIdx.y;

    // Block-uniform parameters -> scalar loads / SALU.
    const float kpx = kpts[(n * NKPT_ + g) * 2 + 0];
    const float kpy = kpts[(n * NKPT_ + g) * 2 + 1];
    const float rl  = roll[n * NKPT_ + g];

    const float kx = kpx * (float)W_;
    const float ky = kpy * (float)H_;
    const bool cond = (rl < P_) && (kx >= 0.0f) && (ky >= 0.0f);

    const float bx = floorf(fmaxf(kx - SHALF_, 0.0f));
    const float ex = floorf(fminf(kx + SHALF_, (float)W_));
    const float by = floorf(fmaxf(ky - SHALF_, 0.0f));
    const float ey = floorf(fminf(ky + SHALF_, (float)H_));

    const int t    = threadIdx.x;
    const int cvec = t & 7;                                  // float4 column in row
    const int row0 = blockIdx.x * ROWS_PER_TILE_ + (t >> 3); // first row, < 192

    // ROW_STRIDE_ = 192 = 2*H_, so h (and thus the whole mask) is loop-invariant.
    const int   h  = row0 % H_;
    const float hf = (float)h;
    const bool inRow = (hf >= by) && (hf < ey);

    bool keep[4];
#pragma unroll
    for (int i = 0; i < 4; ++i) {
        const float wf = (float)(cvec * 4 + i);
        const bool incol = (wf >= bx) && (wf < ex);
        keep[i] = (!cond) || (inRow && incol);
    }
    const bool doLoad = keep[0] | keep[1] | keep[2] | keep[3];

    const size_t slab = ((size_t)(n * C_ + g * CHS_)) * (size_t)(H_ * W_);
    const vf4* src = (const vf4*)(x + slab) + (size_t)row0 * F4_PER_ROW_ + cvec;
    vf4*       dst = (vf4*)(out + slab)     + (size_t)row0 * F4_PER_ROW_ + cvec;

    // Stream prefetch of the far half of this block's range: gets distant
    // lines in flight while the early unrolled loads issue (global_prefetch_b8,
    // speculative -> silently dropped if translation-invalid).
    if (doLoad) {
        __builtin_prefetch(src + (ITERS_ / 2) * F4_STRIDE_, 0, 0);
        __builtin_prefetch(src + (ITERS_ - 1) * F4_STRIDE_, 0, 0);
    }

#pragma unroll 8
    for (int it = 0; it < ITERS_; ++it) {
        vf4 v = (vf4)(0.0f);
        if (doLoad) {
            v = __builtin_nontemporal_load(src);   // global_load_b128 th:NT, EXEC-predicated
        }
        v.x = keep[0] ? v.x : 0.0f;
        v.y = keep[1] ? v.y : 0.0f;
        v.z = keep[2] ? v.z : 0.0f;
        v.w = keep[3] ? v.w : 0.0f;
        __builtin_nontemporal_store(v, dst);       // global_store_b128 th:NT
        src += F4_STRIDE_;
        dst += F4_STRIDE_;
    }
}

extern "C" void kernel_launch(void* const* d_in, const int* in_sizes, int n_in,
                              void* d_out, int out_size, void* d_ws, size_t ws_size,
                              hipStream_t stream) {
    const float* x    = (const float*)d_in[0];  // [64,256,96,32]
    const float* kpts = (const float*)d_in[1];  // [64,17,2]
    const float* roll = (const float*)d_in[2];  // [64,17]
    float* out = (float*)d_out;                 // [64,256,96,32]

    dim3 grid(GRIDX_, GROUPS_, N_);             // 6 x 8 x 64 = 3072 blocks
    dim3 block(256);                            // 8 waves (wave32) per block
    droppart_kernel<<<grid, block, 0, stream>>>(x, kpts, roll, out);
}